// GroundPlane_12197707120720
// MI455X (gfx1250) — compile-verified
//
#include <hip/hip_runtime.h>
#include <hip/hip_bf16.h>

#define HH   384
#define WW   1280
#define HW   (HH * WW)        // 491520
#define GH   192              // int(0.5 * 384)
#define NPTS (GH * WW)        // 245760 ground points per batch
#define NB   4
#define MAXIT 25
#define NUMP  5
#define TOLV  0.1f
#define NPLANES (NB * MAXIT)  // 100
#define TILES_PER_WAVE 60     // 15360 tiles / 256 waves per batch

typedef __attribute__((ext_vector_type(2))) float v2f;
typedef __attribute__((ext_vector_type(8))) float v8f;

// ---------------------------------------------------------------------------
// Kernel 0: zero the per-plane inlier counters (graph-replay safe init)
// ---------------------------------------------------------------------------
__global__ void init_counts(float* __restrict__ counts) {
    if (threadIdx.x < NB * 32) counts[threadIdx.x] = 0.0f;
}

// ---------------------------------------------------------------------------
// Kernel 1: per-plane 3x3 least squares.  ws[i] = inv(A^T A + 1e-6) A^T b
// (note: reference broadcasts +1e-6 onto EVERY entry of A^T A)
// ---------------------------------------------------------------------------
__global__ void solve_planes(const float* __restrict__ pts,
                             const int*  __restrict__ rind,
                             float* __restrict__ wsbuf) {
    int i = blockIdx.x * blockDim.x + threadIdx.x;
    if (i >= NPLANES) return;
    int b = i / MAXIT, m = i % MAXIT;

    const float* bx = pts + (size_t)(b * 3 + 0) * HW + (size_t)GH * WW;
    const float* by = pts + (size_t)(b * 3 + 1) * HW + (size_t)GH * WW;
    const float* bz = pts + (size_t)(b * 3 + 2) * HW + (size_t)GH * WW;

    float S00 = 0.f, S01 = 0.f, S02 = 0.f, S11 = 0.f, S12 = 0.f;
    float t0 = 0.f, t1 = 0.f, t2 = 0.f;
    #pragma unroll
    for (int p = 0; p < NUMP; ++p) {
        int n = rind[b * (MAXIT * NUMP) + m * NUMP + p];
        float x = bx[n], y = by[n], z = bz[n];
        S00 += x * x; S01 += x * z; S02 += x;
        S11 += z * z; S12 += z;
        t0  += x * y; t1  += z * y; t2 += y;
    }
    const float e = 1e-6f;
    float a00 = S00 + e, a01 = S01 + e, a02 = S02 + e;
    float a10 = S01 + e, a11 = S11 + e, a12 = S12 + e;
    float a20 = S02 + e, a21 = S12 + e, a22 = (float)NUMP + e;

    float c00 =  (a11 * a22 - a12 * a21);
    float c01 = -(a10 * a22 - a12 * a20);
    float c02 =  (a10 * a21 - a11 * a20);
    float det = a00 * c00 + a01 * c01 + a02 * c02;
    float inv = 1.0f / det;

    float i00 = c00 * inv;
    float i01 = -(a01 * a22 - a02 * a21) * inv;
    float i02 =  (a01 * a12 - a02 * a11) * inv;
    float i10 = c01 * inv;
    float i11 =  (a00 * a22 - a02 * a20) * inv;
    float i12 = -(a00 * a12 - a02 * a10) * inv;
    float i20 = c02 * inv;
    float i21 = -(a00 * a21 - a01 * a20) * inv;
    float i22 =  (a00 * a11 - a01 * a10) * inv;

    wsbuf[i * 3 + 0] = i00 * t0 + i01 * t1 + i02 * t2;
    wsbuf[i * 3 + 1] = i10 * t0 + i11 * t1 + i12 * t2;
    wsbuf[i * 3 + 2] = i20 * t0 + i21 * t1 + i22 * t2;
}

// ---------------------------------------------------------------------------
// Kernel 2: inlier counting as a WMMA matmul.
//   For point batch c:  D = A(16x4) x W(4x16),  A = [x, z, 1, -y],
//   W columns = [w0, w1, w2, 1]^T  ->  D = w0*x + w1*z + w2 - y  directly.
//   Planes scored against batch c are flat indices i == c (mod 4); slot kk=i/4.
//   A-frag (32-bit 16x4): lanes 0-15 hold K0,K1 (x,z); lanes 16-31 hold K2,K3
//   (1,-y).  C/D: VGPR r = row M (and M+8 in high lanes), lane%16 = col N.
// ---------------------------------------------------------------------------
__global__ void __launch_bounds__(256)
fit_wmma(const float* __restrict__ pts,
         const float* __restrict__ wsbuf,
         float* __restrict__ counts) {
    const int lane = threadIdx.x & 31;
    const int wid  = blockIdx.x * (blockDim.x >> 5) + (threadIdx.x >> 5);
    const int c     = wid >> 8;                 // point batch 0..3 (256 waves each)
    const int tile0 = (wid & 255) * TILES_PER_WAVE;
    const int mrow  = lane & 15;
    const bool hi   = lane >= 16;

    const float* X = pts + (size_t)(c * 3 + 0) * HW + (size_t)GH * WW;
    const float* Y = pts + (size_t)(c * 3 + 1) * HW + (size_t)GH * WW;
    const float* Z = pts + (size_t)(c * 3 + 2) * HW + (size_t)GH * WW;

    // B fragments: group g covers plane slots kk = g*16 + lane%16
    v2f bf0, bf1;
    {
        #pragma unroll
        for (int g = 0; g < 2; ++g) {
            int kk = g * 16 + mrow;
            float w0 = 0.f, w1 = 0.f, w2 = 0.f, w3 = 0.f;
            if (kk < MAXIT) {
                const float* wp = wsbuf + (size_t)(c + 4 * kk) * 3;
                w0 = wp[0]; w1 = wp[1]; w2 = wp[2]; w3 = 1.0f;
            }
            v2f t;
            t.x = hi ? w2 : w0;   // lanes 16-31: K=2, lanes 0-15: K=0
            t.y = hi ? w3 : w1;   // lanes 16-31: K=3, lanes 0-15: K=1
            if (g == 0) bf0 = t; else bf1 = t;
        }
    }

    v8f acc0 = {};
    v8f acc1 = {};
    for (int t = 0; t < TILES_PER_WAVE; ++t) {
        int n = (tile0 + t) * 16 + mrow;
        v2f a;
        if (hi) { a.x = 1.0f; a.y = -Y[n]; }   // K=2 (ones), K=3 (-y)
        else    { a.x = X[n]; a.y =  Z[n]; }   // K=0 (x),    K=1 (z)
        v8f zc = {};
        v8f d0 = __builtin_amdgcn_wmma_f32_16x16x4_f32(
            false, a, false, bf0, (short)0, zc, false, false);
        v8f d1 = __builtin_amdgcn_wmma_f32_16x16x4_f32(
            false, a, false, bf1, (short)0, zc, false, false);
        #pragma unroll
        for (int r = 0; r < 8; ++r) {
            acc0[r] += (fabsf(d0[r]) < TOLV) ? 1.0f : 0.0f;
            acc1[r] += (fabsf(d1[r]) < TOLV) ? 1.0f : 0.0f;
        }
    }
    float s0 = 0.f, s1 = 0.f;
    #pragma unroll
    for (int r = 0; r < 8; ++r) { s0 += acc0[r]; s1 += acc1[r]; }
    // lane l and l+16 both hold column l%16; atomics merge them
    atomicAdd(&counts[c * 32 + mrow], s0);
    atomicAdd(&counts[c * 32 + 16 + mrow], s1);
}

// ---------------------------------------------------------------------------
// Kernel 3: per-batch argmax over 25 iterations (first max wins), emit best_w
// ---------------------------------------------------------------------------
__global__ void pick_best(const float* __restrict__ counts,
                          const float* __restrict__ wsbuf,
                          float* __restrict__ bestw,
                          float* __restrict__ out_tail) {
    int b = threadIdx.x;
    if (b >= NB) return;
    float bestc = -1.0f;
    int besti = b * MAXIT;
    for (int m = 0; m < MAXIT; ++m) {
        int i = b * MAXIT + m;
        float cv = counts[(i & 3) * 32 + (i >> 2)];
        if (cv > bestc) { bestc = cv; besti = i; }
    }
    #pragma unroll
    for (int j = 0; j < 3; ++j) {
        float v = wsbuf[besti * 3 + j];
        bestw[b * 3 + j]    = v;
        out_tail[b * 3 + j] = v;
    }
}

// ---------------------------------------------------------------------------
// Kernel 4: full-image residual, pure streaming, float4 vectorized
// ---------------------------------------------------------------------------
__global__ void __launch_bounds__(256)
compute_dist(const float* __restrict__ pts,
             const float* __restrict__ bestw,
             float* __restrict__ out) {
    int tid = blockIdx.x * blockDim.x + threadIdx.x;   // 0 .. NB*HW/4
    int b   = tid / (HW / 4);
    int p4  = tid % (HW / 4);
    float w0 = bestw[b * 3 + 0];
    float w1 = bestw[b * 3 + 1];
    float w2 = bestw[b * 3 + 2];
    const float4* X = (const float4*)(pts + (size_t)(b * 3 + 0) * HW);
    const float4* Y = (const float4*)(pts + (size_t)(b * 3 + 1) * HW);
    const float4* Z = (const float4*)(pts + (size_t)(b * 3 + 2) * HW);
    float4 x = X[p4], y = Y[p4], z = Z[p4];
    float4 d;
    d.x = fmaf(w0, x.x, fmaf(w1, z.x, w2)) - y.x;
    d.y = fmaf(w0, x.y, fmaf(w1, z.y, w2)) - y.y;
    d.z = fmaf(w0, x.z, fmaf(w1, z.z, w2)) - y.z;
    d.w = fmaf(w0, x.w, fmaf(w1, z.w, w2)) - y.w;
    ((float4*)out)[(size_t)b * (HW / 4) + p4] = d;
}

// ---------------------------------------------------------------------------
extern "C" void kernel_launch(void* const* d_in, const int* in_sizes, int n_in,
                              void* d_out, int out_size, void* d_ws, size_t ws_size,
                              hipStream_t stream) {
    const float* pts  = (const float*)d_in[0];   // (4, 3, 384, 1280) f32
    const int*   rind = (const int*)d_in[1];     // (4, 125) i32
    float* out = (float*)d_out;                  // dist (4*384*1280) ++ best_w (12)
    float* ws  = (float*)d_ws;

    float* wsbuf    = ws;          // 300 floats: 100 planes x 3
    float* counts   = ws + 512;    // 128 floats: 4 batches x 32 slots
    float* bestw    = ws + 768;    // 12 floats
    float* out_tail = out + (size_t)NB * HW;

    hipLaunchKernelGGL(init_counts,  dim3(1),    dim3(128), 0, stream, counts);
    hipLaunchKernelGGL(solve_planes, dim3(1),    dim3(128), 0, stream, pts, rind, wsbuf);
    hipLaunchKernelGGL(fit_wmma,     dim3(128),  dim3(256), 0, stream, pts, wsbuf, counts);
    hipLaunchKernelGGL(pick_best,    dim3(1),    dim3(32),  0, stream, counts, wsbuf, bestw, out_tail);
    hipLaunchKernelGGL(compute_dist, dim3(1920), dim3(256), 0, stream, pts, bestw, out);
}